// FisherRaoDrift_56281251446894
// MI455X (gfx1250) — compile-verified
//
#include <hip/hip_runtime.h>

typedef __attribute__((ext_vector_type(2))) float v2f;
typedef __attribute__((ext_vector_type(8))) float v8f;

static constexpr int KDIM    = 128;
static constexpr int N_EDGES = 25000;   // not derivable from in_sizes; reference constant
static constexpr float EPS   = 1e-12f;

// ---------------- softmax: one wave32 per row, float4 per lane ----------------
__global__ __launch_bounds__(256) void fr_softmax(const float* __restrict__ y,
                                                  float* __restrict__ probs, int n) {
    int wave = threadIdx.x >> 5, lane = threadIdx.x & 31;
    int row = blockIdx.x * 8 + wave;
    if (row >= n) return;
    float4 v = ((const float4*)(y + (size_t)row * KDIM))[lane];
    float m = fmaxf(fmaxf(v.x, v.y), fmaxf(v.z, v.w));
    #pragma unroll
    for (int off = 16; off > 0; off >>= 1) m = fmaxf(m, __shfl_xor(m, off, 32));
    float4 e;
    e.x = __expf(v.x - m); e.y = __expf(v.y - m);
    e.z = __expf(v.z - m); e.w = __expf(v.w - m);
    float s = e.x + e.y + e.z + e.w;
    #pragma unroll
    for (int off = 16; off > 0; off >>= 1) s += __shfl_xor(s, off, 32);
    float inv = 1.f / s;
    e.x *= inv; e.y *= inv; e.z *= inv; e.w *= inv;
    ((float4*)(probs + (size_t)row * KDIM))[lane] = e;
}

// ---------------- degree counts via f32 global atomics ----------------
__global__ __launch_bounds__(256) void fr_degrees(const int* __restrict__ node_idx,
                                                  const int* __restrict__ edge_idx,
                                                  float* __restrict__ edge_deg,
                                                  float* __restrict__ node_deg, int nnz) {
    int i = blockIdx.x * 256 + threadIdx.x;
    if (i >= nnz) return;
    atomicAdd(&edge_deg[edge_idx[i]], 1.0f);
    atomicAdd(&node_deg[node_idx[i]], 1.0f);
}

// ---------------- gather node probs into edge accumulators ----------------
// 32 threads per incidence entry; each thread handles 4 consecutive columns.
__global__ __launch_bounds__(256) void fr_gather(const float* __restrict__ probs,
                                                 const int* __restrict__ node_idx,
                                                 const int* __restrict__ edge_idx,
                                                 float* __restrict__ edge_agg, int nnz) {
    long long gid = (long long)blockIdx.x * 256 + threadIdx.x;
    int i = (int)(gid >> 5);
    if (i >= nnz) return;
    int c4 = (int)(gid & 31);
    int nv = node_idx[i], ev = edge_idx[i];
    float4 p = ((const float4*)(probs + (size_t)nv * KDIM))[c4];
    float* dst = edge_agg + (size_t)ev * KDIM + c4 * 4;
    atomicAdd(dst + 0, p.x); atomicAdd(dst + 1, p.y);
    atomicAdd(dst + 2, p.z); atomicAdd(dst + 3, p.w);
}

// ---------------- edge_feat = (edge_agg / deg) @ W via V_WMMA_F32_16X16X4_F32 ----
// Block: 256 threads = 8 waves. Block owns 16 edge rows; wave w owns cols [16w,16w+16).
// LDS A tile padded to 132 floats/row (stride 132 mod 64 = 4 -> conflict-free strided reads).
__global__ __launch_bounds__(256) void fr_edge_gemm(const float* __restrict__ edge_agg,
                                                    const float* __restrict__ edge_deg,
                                                    const float* __restrict__ W,
                                                    float* __restrict__ edge_feat, int E) {
    __shared__ float sA[16][132];
    int tid = threadIdx.x;
    int m0 = blockIdx.x * 16;
    // coalesced float4 stage of the 16x128 A tile (zero-fill past E)
    for (int i = tid; i < 16 * 32; i += 256) {
        int row = i >> 5, c4 = i & 31, r = m0 + row;
        float4 v = make_float4(0.f, 0.f, 0.f, 0.f);
        if (r < E) v = ((const float4*)edge_agg)[(size_t)r * 32 + c4];
        *(float4*)&sA[row][c4 * 4] = v;
    }
    __syncthreads();

    int wave = tid >> 5, lane = tid & 31;
    int half = lane >> 4, lr = lane & 15;
    int n0 = wave * 16;

    v8f c = {0.f, 0.f, 0.f, 0.f, 0.f, 0.f, 0.f, 0.f};
    #pragma unroll 4
    for (int k0 = 0; k0 < KDIM; k0 += 4) {
        int ka = k0 + 2 * half;                 // lanes 0-15: K=k0,k0+1 ; lanes 16-31: K=k0+2,k0+3
        v2f a, b;
        a.x = sA[lr][ka];
        a.y = sA[lr][ka + 1];
        b.x = W[(size_t)ka * KDIM + n0 + lr];   // B row K=ka, cols n0..n0+15
        b.y = W[(size_t)(ka + 1) * KDIM + n0 + lr];
        c = __builtin_amdgcn_wmma_f32_16x16x4_f32(false, a, false, b, (short)0, c,
                                                  false, false);
    }
    // C layout: VGPR r -> row (8*half + r), col = n0 + lr. Fold edge-mean scale here.
    int rowbase = m0 + 8 * half;
    #pragma unroll
    for (int r = 0; r < 8; ++r) {
        int row = rowbase + r;
        if (row < E) {
            float inv = 1.f / fmaxf(edge_deg[row], 1.f);
            edge_feat[(size_t)row * KDIM + n0 + lr] = c[r] * inv;
        }
    }
}

// ---------------- scatter edge feats back into node accumulators (d_out) ----
__global__ __launch_bounds__(256) void fr_scatter(const float* __restrict__ edge_feat,
                                                  const int* __restrict__ node_idx,
                                                  const int* __restrict__ edge_idx,
                                                  float* __restrict__ out, int nnz) {
    long long gid = (long long)blockIdx.x * 256 + threadIdx.x;
    int i = (int)(gid >> 5);
    if (i >= nnz) return;
    int c4 = (int)(gid & 31);
    int nv = node_idx[i], ev = edge_idx[i];
    float4 f = ((const float4*)(edge_feat + (size_t)ev * KDIM))[c4];
    float* dst = out + (size_t)nv * KDIM + c4 * 4;
    atomicAdd(dst + 0, f.x); atomicAdd(dst + 1, f.y);
    atomicAdd(dst + 2, f.z); atomicAdd(dst + 3, f.w);
}

// ---------------- finalize: nat = clip(p,eps)*acc/deg, subtract row mean -------
__global__ __launch_bounds__(256) void fr_finalize(const float* __restrict__ probs,
                                                   const float* __restrict__ node_deg,
                                                   float* __restrict__ out, int n) {
    int wave = threadIdx.x >> 5, lane = threadIdx.x & 31;
    int row = blockIdx.x * 8 + wave;
    if (row >= n) return;
    float inv = 1.f / fmaxf(node_deg[row], 1.f);
    float4 acc = ((const float4*)(out + (size_t)row * KDIM))[lane];
    float4 p   = ((const float4*)(probs + (size_t)row * KDIM))[lane];
    float4 nat;
    nat.x = fmaxf(p.x, EPS) * acc.x * inv;
    nat.y = fmaxf(p.y, EPS) * acc.y * inv;
    nat.z = fmaxf(p.z, EPS) * acc.z * inv;
    nat.w = fmaxf(p.w, EPS) * acc.w * inv;
    float s = nat.x + nat.y + nat.z + nat.w;
    #pragma unroll
    for (int off = 16; off > 0; off >>= 1) s += __shfl_xor(s, off, 32);
    float mean = s * (1.0f / KDIM);
    nat.x -= mean; nat.y -= mean; nat.z -= mean; nat.w -= mean;
    ((float4*)(out + (size_t)row * KDIM))[lane] = nat;
}

extern "C" void kernel_launch(void* const* d_in, const int* in_sizes, int n_in,
                              void* d_out, int out_size, void* d_ws, size_t ws_size,
                              hipStream_t stream) {
    const float* y        = (const float*)d_in[0];
    const float* W        = (const float*)d_in[1];
    const int*   node_idx = (const int*)d_in[2];
    const int*   edge_idx = (const int*)d_in[3];
    float*       out      = (float*)d_out;

    const int n   = in_sizes[0] / KDIM;   // 100000
    const int nnz = in_sizes[2];          // 1600000
    const int E   = N_EDGES;              // 25000

    // workspace layout (floats)
    float* probs     = (float*)d_ws;
    float* edge_agg  = probs + (size_t)n * KDIM;
    float* edge_feat = edge_agg + (size_t)E * KDIM;
    float* edge_deg  = edge_feat + (size_t)E * KDIM;
    float* node_deg  = edge_deg + E;                  // edge_deg & node_deg contiguous

    // zero accumulators (graph-capture safe)
    hipMemsetAsync(edge_agg, 0, (size_t)E * KDIM * sizeof(float), stream);
    hipMemsetAsync(edge_deg, 0, (size_t)(E + n) * sizeof(float), stream);
    hipMemsetAsync(out, 0, (size_t)n * KDIM * sizeof(float), stream);

    fr_softmax<<<(n + 7) / 8, 256, 0, stream>>>(y, probs, n);
    fr_degrees<<<(nnz + 255) / 256, 256, 0, stream>>>(node_idx, edge_idx,
                                                      edge_deg, node_deg, nnz);
    {
        long long total = (long long)nnz * 32;
        fr_gather<<<(unsigned)((total + 255) / 256), 256, 0, stream>>>(
            probs, node_idx, edge_idx, edge_agg, nnz);
    }
    fr_edge_gemm<<<(E + 15) / 16, 256, 0, stream>>>(edge_agg, edge_deg, W,
                                                    edge_feat, E);
    {
        long long total = (long long)nnz * 32;
        fr_scatter<<<(unsigned)((total + 255) / 256), 256, 0, stream>>>(
            edge_feat, node_idx, edge_idx, out, nnz);
    }
    fr_finalize<<<(n + 7) / 8, 256, 0, stream>>>(probs, node_deg, out, n);
}